// TokenSelection_80771154968642
// MI455X (gfx1250) — compile-verified
//
#include <hip/hip_runtime.h>

typedef __attribute__((ext_vector_type(2))) float v2f;
typedef __attribute__((ext_vector_type(4))) float v4f;
typedef __attribute__((ext_vector_type(8))) float v8f;

#define S_   1024
#define H_   16
#define D_   128
#define NB_  64
#define CBS_ 32
#define KSEL 4

// ---------------------------------------------------------------------------
// Kernel 1: ksum[h][n][d] = sum_c k_compressed[n][h][c][d]
// ---------------------------------------------------------------------------
__global__ void nsa_ksum_kernel(const float* __restrict__ kc,
                                float* __restrict__ ksum) {
    int tid = blockIdx.x * blockDim.x + threadIdx.x;   // over H*NB*D
    if (tid >= H_ * NB_ * D_) return;
    int d = tid % D_;
    int n = (tid / D_) % NB_;
    int h = tid / (D_ * NB_);
    const float* src = kc + (((size_t)n * H_ + h) * CBS_) * D_ + d;
    float acc = 0.0f;
#pragma unroll
    for (int c = 0; c < CBS_; ++c) acc += src[(size_t)c * D_];
    // layout [h][n][d]
    ksum[((size_t)h * NB_ + n) * D_ + d] = acc;
}

// ---------------------------------------------------------------------------
// Kernel 2: per (head, 16-query tile): scores = Q_tile(16x128) x KsumT(128x64)
// via V_WMMA_F32_16X16X4_F32, then top-4 per row -> topidx.
// One wave (32 lanes) per workgroup.
// ---------------------------------------------------------------------------
__global__ void nsa_score_topk_kernel(const float* __restrict__ q,
                                      const float* __restrict__ ksum,
                                      int* __restrict__ topidx) {
    __shared__ float sc[16][68];   // padded: conflict-free spill of accumulators

    const int s0   = blockIdx.x * 16;   // query tile
    const int h    = blockIdx.y;        // head
    const int lane = threadIdx.x;       // 0..31
    const int half = lane >> 4;         // 0: K={k0,k0+1}, 1: K={k0+2,k0+3}
    const int l15  = lane & 15;

    v8f acc0 = {}, acc1 = {}, acc2 = {}, acc3 = {};

    // A fragment: row M = l15 of Q tile, two K values starting at half*2
    const float* qrow  = q    + ((size_t)(s0 + l15) * H_ + h) * D_ + half * 2;
    // B fragment: column N = block index, two K values starting at half*2
    const float* kbase = ksum + ((size_t)h * NB_) * D_ + half * 2;

    for (int k0 = 0; k0 < D_; k0 += 4) {
        v2f a  = *(const v2f*)(qrow + k0);
        v2f b0 = *(const v2f*)(kbase + (size_t)(l15 +  0) * D_ + k0);
        v2f b1 = *(const v2f*)(kbase + (size_t)(l15 + 16) * D_ + k0);
        v2f b2 = *(const v2f*)(kbase + (size_t)(l15 + 32) * D_ + k0);
        v2f b3 = *(const v2f*)(kbase + (size_t)(l15 + 48) * D_ + k0);
        acc0 = __builtin_amdgcn_wmma_f32_16x16x4_f32(false, a, false, b0, (short)0, acc0, false, false);
        acc1 = __builtin_amdgcn_wmma_f32_16x16x4_f32(false, a, false, b1, (short)0, acc1, false, false);
        acc2 = __builtin_amdgcn_wmma_f32_16x16x4_f32(false, a, false, b2, (short)0, acc2, false, false);
        acc3 = __builtin_amdgcn_wmma_f32_16x16x4_f32(false, a, false, b3, (short)0, acc3, false, false);
    }

    // Spill C/D tiles: VGPR v holds row M=v (lanes 0-15) / M=v+8 (lanes 16-31),
    // column N = lane%16 within each 16-wide tile.
#pragma unroll
    for (int v = 0; v < 8; ++v) {
        int row = v + half * 8;
        sc[row][ 0 + l15] = acc0[v];
        sc[row][16 + l15] = acc1[v];
        sc[row][32 + l15] = acc2[v];
        sc[row][48 + l15] = acc3[v];
    }
    __syncthreads();

    // top-4 (descending, ties -> lower index, matching jax.lax.top_k)
    if (lane < 16) {
        float bv0 = -3.4e38f, bv1 = -3.4e38f, bv2 = -3.4e38f, bv3 = -3.4e38f;
        int   bi0 = 0, bi1 = 0, bi2 = 0, bi3 = 0;
        for (int n = 0; n < NB_; ++n) {
            float x = sc[lane][n];
            if (x > bv0)      { bv3=bv2; bi3=bi2; bv2=bv1; bi2=bi1; bv1=bv0; bi1=bi0; bv0=x; bi0=n; }
            else if (x > bv1) { bv3=bv2; bi3=bi2; bv2=bv1; bi2=bi1; bv1=x;   bi1=n; }
            else if (x > bv2) { bv3=bv2; bi3=bi2; bv2=x;   bi2=n; }
            else if (x > bv3) { bv3=x;   bi3=n; }
        }
        int base = ((s0 + lane) * H_ + h) * KSEL;
        topidx[base + 0] = bi0;
        topidx[base + 1] = bi1;
        topidx[base + 2] = bi2;
        topidx[base + 3] = bi3;
    }
}

// ---------------------------------------------------------------------------
// Kernel 3: gather selected blocks. One float4 per thread.
// out_k[s,h,j*CBS+c,d] = kc[idx(s,h,j), h, c, d]   (then same for v)
// Non-temporal stores: 2 GiB output stream must not evict the 32 MiB
// k/v working set from L2 (sources are re-read ~64x each).
// ---------------------------------------------------------------------------
__global__ void nsa_gather_kernel(const float* __restrict__ kc,
                                  const float* __restrict__ vc,
                                  const int* __restrict__ topidx,
                                  float* __restrict__ out) {
    const long long PER4 = (long long)S_ * H_ * KSEL * CBS_ * (D_ / 4); // per-tensor float4 count
    long long tid = (long long)blockIdx.x * blockDim.x + threadIdx.x;
    if (tid >= PER4) return;

    const float* src = (blockIdx.y == 0) ? kc : vc;
    float* dst = out + (blockIdx.y ? PER4 * 4 : 0);

    long long r = tid;
    int d4 = (int)(r & 31); r >>= 5;   // D/4 = 32
    int c  = (int)(r & 31); r >>= 5;   // CBS = 32
    int j  = (int)(r & 3);  r >>= 2;   // KSEL = 4
    int h  = (int)(r & 15); r >>= 4;   // H = 16
    int s  = (int)r;                   // S

    int n = topidx[((s * H_ + h) << 2) + j];

    const v4f val = *(const v4f*)(src + ((((size_t)n * H_ + h) * CBS_ + c) * D_) + (size_t)d4 * 4);
    __builtin_nontemporal_store(val, (v4f*)dst + tid);
}

// ---------------------------------------------------------------------------
extern "C" void kernel_launch(void* const* d_in, const int* in_sizes, int n_in,
                              void* d_out, int out_size, void* d_ws, size_t ws_size,
                              hipStream_t stream) {
    const float* q  = (const float*)d_in[0];   // [1,1024,16,128]
    const float* kc = (const float*)d_in[1];   // [1,64,16,32,128]
    const float* vc = (const float*)d_in[2];   // [1,64,16,32,128]
    float* out = (float*)d_out;                // 2 x [1,1024,16,128,128]

    float* ksum   = (float*)d_ws;                                   // 512 KiB: [16][64][128] f32
    int*   topidx = (int*)((char*)d_ws + (size_t)H_*NB_*D_*sizeof(float)); // 256 KiB

    // 1) reduce k over CBS
    {
        int n = H_ * NB_ * D_;
        nsa_ksum_kernel<<<(n + 255) / 256, 256, 0, stream>>>(kc, ksum);
    }
    // 2) WMMA scoring + top-4 per (s,h)
    {
        dim3 grid(S_ / 16, H_);
        nsa_score_topk_kernel<<<grid, 32, 0, stream>>>(q, ksum, topidx);
    }
    // 3) bulk gather (k and v via grid.y)
    {
        const long long PER4 = (long long)S_ * H_ * KSEL * CBS_ * (D_ / 4);
        dim3 grid((unsigned)((PER4 + 255) / 256), 2);
        nsa_gather_kernel<<<grid, 256, 0, stream>>>(kc, vc, topidx, out);
    }
}